// SimAttention_15032385536419
// MI455X (gfx1250) — compile-verified
//
#include <hip/hip_runtime.h>
#include <hip/hip_bf16.h>
#include <stdint.h>

#define B_ 32
#define N_ 4096
#define D_ 512

// TDM tiling for the key pass: each wave owns 16 rows; K marches in chunks of
// 64 columns. TDM pads each 64-dword row with 4 dwords in LDS -> row stride
// 68 dwords == 4 (mod 64), so ds_load_b64 A-fragment reads are bank-conflict
// free across all 32 lanes.
#define CHUNK      64
#define NCHUNK     (D_ / CHUNK)           // 8
#define ROWPAD_DW  (CHUNK + 4)            // 68 dwords per row in LDS
#define WBUF_BYTES (16 * ROWPAD_DW * 4)   // 4352 B per chunk buffer
#define SMEM_BYTES (8 * 2 * WBUF_BYTES)   // 69632 B: 8 waves, double-buffered

typedef __attribute__((ext_vector_type(2))) float v2f;
typedef __attribute__((ext_vector_type(4))) float v4f;
typedef __attribute__((ext_vector_type(8))) float v8f;
typedef __attribute__((ext_vector_type(4))) unsigned int v4u;
typedef __attribute__((ext_vector_type(4))) int v4i;
typedef __attribute__((ext_vector_type(8))) int v8i;

#if defined(__AMDGCN__) && __has_builtin(__builtin_amdgcn_tensor_load_to_lds)
#define USE_TDM 1
#else
#define USE_TDM 0
#endif

#if USE_TDM
// ---------------------------------------------------------------------------
// Issue one TDM load of a 16-row x CHUNK-col f32 tile (row stride D_ elems in
// memory) into LDS at lds_addr, with 4-dword padding after every CHUNK dwords.
// D# layout per CDNA5 ISA ch.8: group0 = {count/type/addrs}, group1 =
// {flags/dims/strides}; groups 2-3 zero (2-D tensor). Wave-uniform args.
// ---------------------------------------------------------------------------
__device__ __forceinline__ void tdm_load_tile(const float* gsrc,
                                              unsigned lds_addr) {
  const unsigned long long ga = (unsigned long long)(uintptr_t)gsrc;

  v4u g0;
  g0.x = 1u;                                   // count=1 (valid user D#)
  g0.y = lds_addr;                             // lds_addr [63:32]
  g0.z = (unsigned)(ga & 0xffffffffu);         // global_addr[31:0]
  g0.w = (unsigned)((ga >> 32) & 0x1ffffffu)   // global_addr[56:32]
         | (2u << 30);                         // type=2 ("image")

  v8i g1;
  g1.s0 = (int)((2u << 16)                     // data_size = 4 B
                | (1u << 20)                   // pad_enable
                | (5u << 22)                   // pad_interval: 2^(5+1)=64 dw
                | (3u << 25));                 // pad_amount: 3+1 = 4 dwords
  g1.s1 = (int)((unsigned)CHUNK << 16);        // tensor_dim0[15:0] @bits63:48
  g1.s2 = (int)(16u << 16);                    // tensor_dim1[15:0] @bits95:80
  g1.s3 = (int)((unsigned)CHUNK << 16);        // tile_dim0 @bits127:112
  g1.s4 = (int)16u;                            // tile_dim1=16, tile_dim2=0
  g1.s5 = (int)(unsigned)D_;                   // tensor_dim0_stride[31:0]
  g1.s6 = 0;                                   // stride0[47:32], stride1[15:0]
  g1.s7 = 0;                                   // stride1[47:16]

  v4i z4 = {0, 0, 0, 0};
#if __clang_major__ >= 23
  v8i z8 = {0, 0, 0, 0, 0, 0, 0, 0};
  __builtin_amdgcn_tensor_load_to_lds(g0, g1, z4, z4, z8, 0);
#else
  __builtin_amdgcn_tensor_load_to_lds(g0, g1, z4, z4, 0);
#endif
}
#endif  // USE_TDM

// ---------------------------------------------------------------------------
// One chunk of the fused K.q WMMA + sum-of-squares, A-fragments from LDS.
//   A (16x4 f32): lane (m = L&15, hi = L>>4) reads K[row m, 4s+2hi .. +1]
//   B (4x16 f32): q[4s+k] replicated across columns (same per-lane load)
// ---------------------------------------------------------------------------
__device__ __forceinline__ void chunk_mma(const char* __restrict__ lbuf,
                                          const float* __restrict__ qc,
                                          int m, int hi, v8f& acc, float& ss) {
  const float* lrow = (const float*)lbuf + m * ROWPAD_DW + 2 * hi;
  const float* qp   = qc + 2 * hi;
#pragma unroll 8
  for (int s = 0; s < CHUNK / 4; ++s) {
    v2f a  = *(const v2f*)(lrow + 4 * s);
    v2f bb = *(const v2f*)(qp + 4 * s);
    ss += a.x * a.x + a.y * a.y;
    acc = __builtin_amdgcn_wmma_f32_16x16x4_f32(
        false, a, false, bb, (short)0, acc, false, false);
  }
}

// ---------------------------------------------------------------------------
// Kernel 1: dot[b,n] = K.q via v_wmma_f32_16x16x4_f32, plus ssq[b,n] = |k_n|^2.
// Grid (N/128, B), block 256 (8 waves x 16 rows). TDM double-buffer pipeline:
// tensor_load_to_lds streams each 16x64 K tile (bulk sequential HBM traffic)
// while the wave WMMAs on the other buffer; s_wait_tensorcnt paces the pipe.
// EXEC stays all-ones through the WMMA loop (ISA requirement).
// ---------------------------------------------------------------------------
__global__ __launch_bounds__(256) void k_dot_norm(
    const float* __restrict__ q, const float* __restrict__ key,
    float* __restrict__ dot, float* __restrict__ ssq) {
  extern __shared__ char smem[];
  const int b    = blockIdx.y;
  const int wave = threadIdx.x >> 5;
  const int lane = threadIdx.x & 31;
  const int m    = lane & 15;
  const int hi   = lane >> 4;
  const int n0   = blockIdx.x * 128 + wave * 16;

  const float* krow = key + ((size_t)b * N_ + n0) * D_;  // wave's 16-row tile
  const float* qb   = q + (size_t)b * D_;

  v8f acc = {};
  float ss = 0.f;

#if USE_TDM
  const unsigned buf0 = (unsigned)(uintptr_t)smem + wave * 2 * WBUF_BYTES;
  const unsigned buf1 = buf0 + WBUF_BYTES;
  const char* lb0 = smem + wave * 2 * WBUF_BYTES;
  const char* lb1 = lb0 + WBUF_BYTES;

  tdm_load_tile(krow + 0 * CHUNK, buf0);
  tdm_load_tile(krow + 1 * CHUNK, buf1);

#pragma unroll
  for (int c = 0; c < NCHUNK; ++c) {
    if (c + 1 < NCHUNK) {
      __builtin_amdgcn_s_wait_tensorcnt(1);  // chunk c landed, c+1 in flight
    } else {
      __builtin_amdgcn_s_wait_tensorcnt(0);  // last chunk landed
    }
    chunk_mma((c & 1) ? lb1 : lb0, qb + c * CHUNK, m, hi, acc, ss);
    if (c + 2 < NCHUNK) {
      // ds reads of this buffer are complete (their results fed the WMMAs
      // above), so it is safe to overwrite it with chunk c+2.
      tdm_load_tile(krow + (c + 2) * CHUNK, (c & 1) ? buf1 : buf0);
    }
  }
#else
  // Fallback: direct global A-fragment loads (previous round's proven path).
  const float* kp = krow + (size_t)m * D_ + 2 * hi;
  const float* qp = qb + 2 * hi;
#pragma unroll 4
  for (int s = 0; s < D_ / 4; ++s) {
    v2f a  = *(const v2f*)(kp + 4 * s);
    v2f bb = *(const v2f*)(qp + 4 * s);
    ss += a.x * a.x + a.y * a.y;
    acc = __builtin_amdgcn_wmma_f32_16x16x4_f32(
        false, a, false, bb, (short)0, acc, false, false);
  }
#endif

  // Row m sum-of-squares = lane m (cols 4s..4s+1) + lane m+16 (cols 4s+2..+3).
  ss += __shfl_xor(ss, 16, 32);
  if (lane < 16) ssq[(size_t)b * N_ + n0 + lane] = ss;

  // C/D layout: lane 0 VGPR r -> D[r,0] (rows 0..7); lane 16 -> rows 8..15.
  if (lane == 0 || lane == 16) {
    float* dp = dot + (size_t)b * N_ + n0 + hi * 8;
    v4f lo = {acc.s0, acc.s1, acc.s2, acc.s3};
    v4f hh = {acc.s4, acc.s5, acc.s6, acc.s7};
    *(v4f*)(dp)     = lo;
    *(v4f*)(dp + 4) = hh;
  }
}

// ---------------------------------------------------------------------------
// Kernel 2: per-batch normalization (one block per b).
//   attn = dot / max(|q|*|k_n|, 1e-8); masked; thresholded at 0.9;
//   attn /= (sum + 1e-8). Overwrites `dot` in place with attn.
// ---------------------------------------------------------------------------
__global__ __launch_bounds__(256) void k_attn(
    const float* __restrict__ q, const uint8_t* __restrict__ mask,
    float* __restrict__ dot, const float* __restrict__ ssq) {
  const int b   = blockIdx.x;
  const int tid = threadIdx.x;
  __shared__ float red[256];
  __shared__ float s_bcast;

  float t = 0.f;
  for (int d = tid; d < D_; d += 256) {
    float v = q[(size_t)b * D_ + d];
    t += v * v;
  }
  red[tid] = t;
  __syncthreads();
  for (int off = 128; off > 0; off >>= 1) {
    if (tid < off) red[tid] += red[tid + off];
    __syncthreads();
  }
  if (tid == 0) s_bcast = sqrtf(red[0]);
  __syncthreads();
  const float normq = s_bcast;
  __syncthreads();

  float psum = 0.f;
  for (int n = tid; n < N_; n += 256) {
    const size_t idx = (size_t)b * N_ + n;
    float denom = fmaxf(normq * sqrtf(ssq[idx]), 1e-8f);
    float c = dot[idx] / denom;
    if (mask[idx]) c = 0.f;
    if (c < 0.9f) c = 0.f;
    psum += c;
    dot[idx] = c;
  }
  red[tid] = psum;
  __syncthreads();
  for (int off = 128; off > 0; off >>= 1) {
    if (tid < off) red[tid] += red[tid + off];
    __syncthreads();
  }
  if (tid == 0) s_bcast = 1.f / (red[0] + 1e-8f);
  __syncthreads();
  const float inv = s_bcast;

  for (int n = tid; n < N_; n += 256) dot[(size_t)b * N_ + n] *= inv;
}

// ---------------------------------------------------------------------------
// Kernel 3: out[b,d] = sum_n attn[b,n] * value[b,n,d].
// One block per b; thread owns a float2 of d (256 x 8 B = fully coalesced 2 KB
// per visited row). attn staged in LDS; zero-weight rows skipped via a
// block-uniform branch -> ~99.8% of the 256 MB value traffic avoided when the
// 0.9 threshold is sparse, while staying correct in general.
// ---------------------------------------------------------------------------
__global__ __launch_bounds__(256) void k_out(
    const float* __restrict__ attn, const float* __restrict__ value,
    float* __restrict__ out) {
  const int b   = blockIdx.x;
  const int tid = threadIdx.x;
  __shared__ float s_attn[N_];

  for (int n = tid; n < N_; n += 256) s_attn[n] = attn[(size_t)b * N_ + n];
  __syncthreads();

  v2f acc = {0.f, 0.f};
  const float* vp = value + (size_t)b * N_ * D_ + 2 * tid;
  for (int n = 0; n < N_; ++n) {
    const float a = s_attn[n];  // uniform across the block
    if (a != 0.f) {
      v2f v = *(const v2f*)(vp + (size_t)n * D_);
      acc.x += a * v.x;
      acc.y += a * v.y;
    }
  }
  *(v2f*)(out + (size_t)b * D_ + 2 * tid) = acc;
}

// ---------------------------------------------------------------------------
extern "C" void kernel_launch(void* const* d_in, const int* in_sizes, int n_in,
                              void* d_out, int out_size, void* d_ws,
                              size_t ws_size, hipStream_t stream) {
  const float*   q    = (const float*)d_in[0];    // [B,D]
  const float*   key  = (const float*)d_in[1];    // [B,N,D]
  const float*   val  = (const float*)d_in[2];    // [B,N,D]
  const uint8_t* mask = (const uint8_t*)d_in[3];  // [B,N] bool (1 byte)
  float*         out  = (float*)d_out;            // [B,D]

  float* dot = (float*)d_ws;           // [B,N], later holds attn
  float* ssq = dot + (size_t)B_ * N_;  // [B,N]

  // Allow >64KB dynamic LDS for the TDM double buffers (idempotent).
  (void)hipFuncSetAttribute((const void*)k_dot_norm,
                            hipFuncAttributeMaxDynamicSharedMemorySize,
                            SMEM_BYTES);

  dim3 g1(N_ / 128, B_);
  k_dot_norm<<<g1, 256, SMEM_BYTES, stream>>>(q, key, dot, ssq);

  k_attn<<<B_, 256, 0, stream>>>(q, mask, dot, ssq);

  k_out<<<B_, 256, 0, stream>>>(dot, val, out);
}